// FPLPGCN_dw_1168231104595
// MI455X (gfx1250) — compile-verified
//
#include <hip/hip_runtime.h>
#include <hip/hip_bf16.h>
#include <stdint.h>

typedef __attribute__((ext_vector_type(2))) float v2f;
typedef __attribute__((ext_vector_type(8))) float v8f;

static inline int cdiv(long long a, long long b) { return (int)((a + b - 1) / b); }

// Guaranteed single HW instruction: non-returning fp32 global atomic at device scope.
__device__ __forceinline__ void atom_add_f32(float* p, float v) {
  asm volatile("global_atomic_add_f32 %0, %1, off scope:SCOPE_DEV"
               :: "v"(p), "v"(v) : "memory");
}

// ---------------- elementwise helpers ----------------

__global__ __launch_bounds__(256) void k_fill(float* p, int n, float v) {
  int i = blockIdx.x * 256 + threadIdx.x;
  if (i < n) p[i] = v;
}

__global__ __launch_bounds__(256) void k_deg(float* deg, const long long* __restrict__ col, int E) {
  int e = blockIdx.x * 256 + threadIdx.x;
  if (e < E) atom_add_f32(&deg[(int)col[e]], 1.0f);
}

__global__ __launch_bounds__(256) void k_rsqrt(float* p, int n) {
  int i = blockIdx.x * 256 + threadIdx.x;
  if (i < n) p[i] = rsqrtf(p[i]);   // deg >= 1 always (self loop)
}

template <int F>
__global__ __launch_bounds__(256) void k_fill_bias(float* __restrict__ out,
                                                   const float* __restrict__ bias, int n) {
  int i = blockIdx.x * 256 + threadIdx.x;
  if (i < n) out[i] = bias[i & (F - 1)];
}

__global__ __launch_bounds__(256) void k_relu(float* p, int n) {
  int i = blockIdx.x * 256 + threadIdx.x;
  if (i < n) p[i] = fmaxf(p[i], 0.0f);
}

__global__ __launch_bounds__(256) void k_label_epi(const float* __restrict__ ag,
                                                   const float* __restrict__ y,
                                                   const unsigned char* __restrict__ mask,
                                                   float* __restrict__ xl, int n16, int do_relu) {
  int i = blockIdx.x * 256 + threadIdx.x;
  if (i >= n16) return;
  int node = i >> 4;
  float v = ag[i];
  if (do_relu) v = fmaxf(v, 0.0f);
  xl[i] = mask[node] ? y[i] : v;
}

// ---------------- WMMA fp32 GEMM: C[M x NOUT] = A[M x K] @ B[K x NOUT] ----------------
// One wave computes one 16x16 output tile; K stepped by 4 using V_WMMA_F32_16X16X4_F32.
// fp32 A 16x4 layout: lane(0-15)=M, vgpr v & lane-half h -> K = 2*h + v  (v2f per lane)
// fp32 B 4x16 layout: lane%16=N,  K = 2*(lane/16) + v
// fp32 C/D: vgpr j, lane -> M = j + 8*(lane/16), N = lane%16

template <int K, int NOUT>
__global__ __launch_bounds__(256) void k_gemm(const float* __restrict__ A,
                                              const float* __restrict__ B,
                                              float* __restrict__ C, int M) {
  int wave = (blockIdx.x * 256 + threadIdx.x) >> 5;
  int lane = threadIdx.x & 31;
  constexpr int TN = NOUT / 16;
  int tm = (wave / TN) * 16;
  int tn = (wave % TN) * 16;
  if (tm >= M) return;           // whole-wave uniform exit
  int half = lane >> 4;
  int l = lane & 15;
  const float* arow = A + (size_t)(tm + l) * K;
  v8f c = {};
#pragma unroll
  for (int k = 0; k < K; k += 4) {
    int ka = k + 2 * half;
    v2f a = *(const v2f*)(arow + ka);
    v2f b;
    b.x = B[ka * NOUT + tn + l];
    b.y = B[(ka + 1) * NOUT + tn + l];
    c = __builtin_amdgcn_wmma_f32_16x16x4_f32(false, a, false, b, (short)0, c, false, false);
  }
#pragma unroll
  for (int j = 0; j < 8; j++)
    C[(size_t)(tm + j + 8 * half) * NOUT + tn + l] = c[j];
}

// ---------------- edge scatter: out[col] += src[row] * norm  (plus self loops) ----------------
// TPE threads cooperate on one edge; feature rows are contiguous -> coalesced loads & atomics.

template <int F, int TPE>
__global__ __launch_bounds__(256) void k_aggregate(const float* __restrict__ src,
                                                   float* __restrict__ out,
                                                   const long long* __restrict__ row,
                                                   const long long* __restrict__ col,
                                                   const float* __restrict__ dis,
                                                   int E, int N) {
  long long gid = (long long)blockIdx.x * 256 + threadIdx.x;
  int e = (int)(gid / TPE);
  int f0 = (int)(gid % TPE);
  if (e >= E + N) return;
  int r, cn;
  float nrm;
  if (e < E) {
    r = (int)row[e];
    cn = (int)col[e];
    nrm = dis[r] * dis[cn];
  } else {               // self loop
    r = cn = e - E;
    float d = dis[r];
    nrm = d * d;
  }
  const float* s = src + (size_t)r * F;
  float* o = out + (size_t)cn * F;
#pragma unroll
  for (int f = f0; f < F; f += TPE) atom_add_f32(o + f, s[f] * nrm);
}

// ---------------- fused head: sigmoid([h | xl | dw] @ Wf + bf) ----------------
// K = 208 = 128 + 16 + 64; all boundaries are multiples of 4 so each K-step reads one source.

__global__ __launch_bounds__(256) void k_fuse(const float* __restrict__ h,
                                              const float* __restrict__ xl,
                                              const float* __restrict__ dwe,
                                              const float* __restrict__ Wf,
                                              const float* __restrict__ bf,
                                              float* __restrict__ out, int M) {
  int wave = (blockIdx.x * 256 + threadIdx.x) >> 5;
  int lane = threadIdx.x & 31;
  int tm = wave * 16;
  if (tm >= M) return;
  int half = lane >> 4;
  int l = lane & 15;
  int rowi = tm + l;
  v8f c = {};
#pragma unroll
  for (int k = 0; k < 208; k += 4) {
    int ka = k + 2 * half;
    v2f a, b;
    if (k < 128)
      a = *(const v2f*)(h + (size_t)rowi * 128 + ka);
    else if (k < 144)
      a = *(const v2f*)(xl + (size_t)rowi * 16 + (ka - 128));
    else
      a = *(const v2f*)(dwe + (size_t)rowi * 64 + (ka - 144));
    b.x = Wf[ka * 16 + l];
    b.y = Wf[(ka + 1) * 16 + l];
    c = __builtin_amdgcn_wmma_f32_16x16x4_f32(false, a, false, b, (short)0, c, false, false);
  }
  float bb = bf[l];
#pragma unroll
  for (int j = 0; j < 8; j++) {
    float v = c[j] + bb;
    out[(size_t)(tm + j + 8 * half) * 16 + l] = 1.0f / (1.0f + __expf(-v));
  }
}

// ---------------- host side ----------------

extern "C" void kernel_launch(void* const* d_in, const int* in_sizes, int n_in,
                              void* d_out, int out_size, void* d_ws, size_t ws_size,
                              hipStream_t stream) {
  const float* x   = (const float*)d_in[0];         // N x 128
  const float* y   = (const float*)d_in[1];         // N x 16
  const long long* ei = (const long long*)d_in[2];  // 2 x E (int64)
  const float* dwe = (const float*)d_in[3];         // N x 64
  const unsigned char* mask = (const unsigned char*)d_in[4];  // N (bool)
  const float* W1 = (const float*)d_in[5];          // 128 x 128
  const float* b1 = (const float*)d_in[6];          // 128
  const float* W2 = (const float*)d_in[7];          // 128 x 128
  const float* b2 = (const float*)d_in[8];          // 128
  const float* Wl = (const float*)d_in[9];          // 10 x 16 x 16
  const float* bl = (const float*)d_in[10];         // 10 x 16
  const float* Wf = (const float*)d_in[11];         // 208 x 16
  const float* bf = (const float*)d_in[12];         // 16
  float* out = (float*)d_out;

  const int N = in_sizes[0] / 128;
  const int E = in_sizes[2] / 2;
  const long long* row = ei;
  const long long* col = ei + E;

  float* ws  = (float*)d_ws;
  float* dis = ws;                          // N
  float* t   = dis + N;                     // N x 128 (gemm output scratch)
  float* h   = t + (size_t)N * 128;         // N x 128 (layer1 out / layer2 in)
  float* h2  = h + (size_t)N * 128;         // N x 128 (layer2 out)
  float* xl  = h2 + (size_t)N * 128;        // N x 16  (label state)
  float* tl  = xl + (size_t)N * 16;         // N x 16  (label gemm out)
  float* ag  = tl + (size_t)N * 16;         // N x 16  (label aggregate)

  const int tilesM = N / 16;                // N = 100000 -> 6250 exact

  // --- degree / symmetric norm (self-loop folded into init = 1.0) ---
  k_fill<<<cdiv(N, 256), 256, 0, stream>>>(dis, N, 1.0f);
  k_deg<<<cdiv(E, 256), 256, 0, stream>>>(dis, col, E);
  k_rsqrt<<<cdiv(N, 256), 256, 0, stream>>>(dis, N);

  // --- GCN layer 1: relu(scatter(x @ W1) + b1) ---
  k_gemm<128, 128><<<cdiv((long long)tilesM * 8 * 32, 256), 256, 0, stream>>>(x, W1, t, N);
  k_fill_bias<128><<<cdiv((long long)N * 128, 256), 256, 0, stream>>>(h, b1, N * 128);
  k_aggregate<128, 32><<<cdiv((long long)(E + N) * 32, 256), 256, 0, stream>>>(t, h, row, col, dis, E, N);
  k_relu<<<cdiv((long long)N * 128, 256), 256, 0, stream>>>(h, N * 128);

  // --- GCN layer 2: scatter(h @ W2) + b2 ---
  k_gemm<128, 128><<<cdiv((long long)tilesM * 8 * 32, 256), 256, 0, stream>>>(h, W2, t, N);
  k_fill_bias<128><<<cdiv((long long)N * 128, 256), 256, 0, stream>>>(h2, b2, N * 128);
  k_aggregate<128, 32><<<cdiv((long long)(E + N) * 32, 256), 256, 0, stream>>>(t, h2, row, col, dis, E, N);

  // --- label propagation: 10 rounds of 16-wide GCN conv with mask reset ---
  for (int j = 0; j < 10; j++) {
    const float* xin = (j == 0) ? y : xl;
    k_gemm<16, 16><<<cdiv((long long)tilesM * 32, 256), 256, 0, stream>>>(xin, Wl + j * 256, tl, N);
    k_fill_bias<16><<<cdiv((long long)N * 16, 256), 256, 0, stream>>>(ag, bl + j * 16, N * 16);
    k_aggregate<16, 16><<<cdiv((long long)(E + N) * 16, 256), 256, 0, stream>>>(tl, ag, row, col, dis, E, N);
    k_label_epi<<<cdiv((long long)N * 16, 256), 256, 0, stream>>>(ag, y, mask, xl, N * 16, j != 9);
  }

  // --- fused head ---
  k_fuse<<<cdiv((long long)tilesM * 32, 256), 256, 0, stream>>>(h2, xl, dwe, Wf, bf, out, N);
}